// ROIPooler_52956946760068
// MI455X (gfx1250) — compile-verified
//
#include <hip/hip_runtime.h>
#include <hip/hip_bf16.h>

typedef float v2f __attribute__((ext_vector_type(2)));
typedef float v8f __attribute__((ext_vector_type(8)));

#define OUTSZ 7
#define XK 40   // patch cols = K of stage 1 (10 x WMMA k=4)
#define XP 42   // padded LDS row length for patch / Ax (42*l mod 64 distinct -> no bank conflicts)
#define YK 32   // patch rows = K of stage 2 (8 x WMMA k=4)
#define YP 33   // padded LDS row length for Ay / T'

__device__ __forceinline__ int imin(int a, int b) { return a < b ? a : b; }
__device__ __forceinline__ int imax(int a, int b) { return a > b ? a : b; }

__device__ __forceinline__ v8f wmma_f32_16x16x4(v2f a, v2f b, v8f c) {
  // 8-arg f32-class signature: (neg_a, A, neg_b, B, c_mod, C, reuse_a, reuse_b)
  return __builtin_amdgcn_wmma_f32_16x16x4_f32(false, a, false, b, (short)0, c,
                                               false, false);
}

__global__ __launch_bounds__(256)
void roi_align_wmma(const float* __restrict__ f2, const float* __restrict__ f3,
                    const float* __restrict__ f4, const float* __restrict__ f5,
                    const float* __restrict__ boxes, const int* __restrict__ bidx,
                    float* __restrict__ out)
{
  __shared__ float sAy[16][YP];        // y-axis weights: out-row x patch-row
  __shared__ float sAx[16][XP];        // x-axis weights: out-col x patch-col
  __shared__ float sP [8][YK][XP];     // per-wave feature patch
  __shared__ float sT [8][16][YP];     // per-wave T transposed: [n][r]
  __shared__ int   sI[5];              // H, W, ybase, xbase, lvl
  __shared__ long long sOff;           // batch plane offset (elements)

  const int m   = blockIdx.x;
  const int tid = threadIdx.x;

  // Zero the weight matrices (padded regions must multiply to 0).
  for (int i = tid; i < 16 * YP; i += 256) (&sAy[0][0])[i] = 0.0f;
  for (int i = tid; i < 16 * XP; i += 256) (&sAx[0][0])[i] = 0.0f;
  __syncthreads();

  if (tid == 0) {
    const float x0 = boxes[4 * m + 0], y0b = boxes[4 * m + 1];
    const float x1 = boxes[4 * m + 2], y1  = boxes[4 * m + 3];
    // --- level assignment ---
    const float size = sqrtf((x1 - x0) * (y1 - y0b));
    float lf = floorf(4.0f + log2f(size * (1.0f / 224.0f) + 1e-8f));
    lf = fminf(fmaxf(lf, 2.0f), 5.0f);
    const int lvl = (int)lf - 2;
    const int   Hs[4] = {200, 100, 50, 25};
    const int   Ws[4] = {304, 152, 76, 38};
    const float Ss[4] = {0.25f, 0.125f, 0.0625f, 0.03125f};
    const int H = Hs[lvl], W = Ws[lvl];
    const float sc = Ss[lvl];
    sOff = (long long)bidx[m] * 256 * (long long)(H * W);

    const float bx0 = x0 * sc - 0.5f, by0 = y0b * sc - 0.5f;
    const float bw  = (x1 * sc - 0.5f - bx0) * (1.0f / OUTSZ);
    const float bh  = (y1 * sc - 0.5f - by0) * (1.0f / OUTSZ);

    // --- patch origin: floor index of first sample per axis (samples are monotone) ---
    int yb, xb;
    {
      float cc  = fmaxf(by0 + bh * 0.25f, 0.0f);
      float lof = floorf(cc);
      int lo = (lof >= (float)(H - 1)) ? (H - 1) : (int)lof;
      yb = imin(lo, imax(0, H - YK));            // keep 32-row window near-valid
    }
    {
      float cc  = fmaxf(bx0 + bw * 0.25f, 0.0f);
      float lof = floorf(cc);
      int lo = (lof >= (float)(W - 1)) ? (W - 1) : (int)lof;
      xb = imin(lo, imax(0, W - XK)) & ~3;       // 16B-align left edge
    }
    sI[0] = H; sI[1] = W; sI[2] = yb; sI[3] = xb; sI[4] = lvl;

    // --- scatter bilinear weights into Ay / Ax (includes the 2x2-avg 0.5 factors) ---
    for (int s = 0; s < 14; ++s) {
      const float off = (float)(s >> 1) + ((s & 1) ? 0.75f : 0.25f);
      const int o = s >> 1;
      { // y axis
        const float c  = by0 + bh * off;
        const float vd = (c >= -1.0f && c <= (float)H) ? 1.0f : 0.0f;
        const float cc = fmaxf(c, 0.0f);
        const float lof = floorf(cc);
        const bool cap = lof >= (float)(H - 1);
        const int lo = cap ? H - 1 : (int)lof;
        const int hi = cap ? H - 1 : lo + 1;
        const float l = cap ? 0.0f : (cc - lof);
        sAy[o][imin(imax(lo - yb, 0), YK - 1)] += 0.5f * (1.0f - l) * vd;
        sAy[o][imin(imax(hi - yb, 0), YK - 1)] += 0.5f * l * vd;
      }
      { // x axis
        const float c  = bx0 + bw * off;
        const float vd = (c >= -1.0f && c <= (float)W) ? 1.0f : 0.0f;
        const float cc = fmaxf(c, 0.0f);
        const float lof = floorf(cc);
        const bool cap = lof >= (float)(W - 1);
        const int lo = cap ? W - 1 : (int)lof;
        const int hi = cap ? W - 1 : lo + 1;
        const float l = cap ? 0.0f : (cc - lof);
        sAx[o][imin(imax(lo - xb, 0), XK - 1)] += 0.5f * (1.0f - l) * vd;
        sAx[o][imin(imax(hi - xb, 0), XK - 1)] += 0.5f * l * vd;
      }
    }
  }
  __syncthreads();

  const int H  = sI[0], W = sI[1], yb = sI[2], xb = sI[3];
  const int lvl = sI[4];
  // Reselect the kernarg pointer so address-space inference keeps these as
  // GLOBAL loads (storing the pointer through LDS forced flat_load last round).
  const float* __restrict__ fsel =
      (lvl == 0) ? f2 : (lvl == 1) ? f3 : (lvl == 2) ? f4 : f5;
  const float* __restrict__ fbase = fsel + (size_t)sOff;

  const int w    = tid >> 5;
  const int lane = tid & 31;
  const int hi16 = lane >> 4;   // WMMA lane half: K offset {0,1} vs {2,3}
  const int mm   = lane & 15;   // WMMA row / column within half

  const int rr = imin(yb + lane, H - 1);                 // this lane's patch row (clamped)
  const size_t plane  = (size_t)(H * W);
  const size_t rowoff = (size_t)rr * W + xb;
  const int    cmax   = W - 2 - xb;                      // clamp for in-row b64 pairs

  for (int ci = 0; ci < 32; ++ci) {
    const int c = ci * 8 + w;                            // channel handled by this wave
    const float* rowp = fbase + (size_t)c * plane + rowoff;

    // prefetch next channel's patch row (global_prefetch_b8)
    if (ci < 31)
      __builtin_prefetch(rowp + 8 * plane, 0, 3);

    // --- batch-load one contiguous patch row per lane (20 pipelined b64 loads),
    //     then commit to LDS; keeps a single loadcnt wait for the whole batch ---
    v2f tmp[XK / 2];
    #pragma unroll
    for (int j = 0; j < XK / 2; ++j)
      tmp[j] = *(const v2f*)(rowp + imin(2 * j, cmax));
    #pragma unroll
    for (int j = 0; j < XK / 2; ++j) {
      sP[w][lane][2 * j]     = tmp[j].x;
      sP[w][lane][2 * j + 1] = tmp[j].y;
    }
    __asm__ volatile("s_wait_dscnt 0" ::: "memory");     // LDS writes visible to wave

    // --- stage 1: T(32x16) = P(32x40) * Ax^T(40x16), two 16-row halves ---
    #pragma unroll
    for (int h = 0; h < 2; ++h) {
      v8f acc = {};
      #pragma unroll
      for (int k = 0; k < XK / 4; ++k) {
        const int kk = 4 * k + 2 * hi16;
        v2f a; a.x = sP[w][16 * h + mm][kk]; a.y = sP[w][16 * h + mm][kk + 1];
        v2f b; b.x = sAx[mm][kk];            b.y = sAx[mm][kk + 1];
        acc = wmma_f32_16x16x4(a, b, acc);
      }
      #pragma unroll
      for (int v = 0; v < 8; ++v)                        // D layout -> T' [n][r]
        sT[w][mm][16 * h + v + 8 * hi16] = acc[v];
    }
    __asm__ volatile("s_wait_dscnt 0" ::: "memory");

    // --- stage 2: O(16x16) = Ay(16x32) * T(32x16) ---
    v8f o = {};
    #pragma unroll
    for (int k = 0; k < YK / 4; ++k) {
      const int kk = 4 * k + 2 * hi16;
      v2f a; a.x = sAy[mm][kk];    a.y = sAy[mm][kk + 1];
      v2f b; b.x = sT[w][mm][kk];  b.y = sT[w][mm][kk + 1];
      o = wmma_f32_16x16x4(a, b, o);
    }

    // --- store valid 7x7 from D layout ---
    if (mm < OUTSZ) {
      float* op = out + ((size_t)m * 256 + c) * (OUTSZ * OUTSZ) + mm;
      #pragma unroll
      for (int v = 0; v < 8; ++v) {
        const int oy = v + 8 * hi16;
        if (oy < OUTSZ) op[oy * OUTSZ] = o[v];
      }
    }
  }
}

extern "C" void kernel_launch(void* const* d_in, const int* in_sizes, int n_in,
                              void* d_out, int out_size, void* d_ws, size_t ws_size,
                              hipStream_t stream) {
  (void)n_in; (void)out_size; (void)d_ws; (void)ws_size;
  const float* f2    = (const float*)d_in[0];
  const float* f3    = (const float*)d_in[1];
  const float* f4    = (const float*)d_in[2];
  const float* f5    = (const float*)d_in[3];
  const float* boxes = (const float*)d_in[4];
  const int*   bidx  = (const int*)d_in[5];
  const int M = in_sizes[4] / 4;   // 1000 boxes
  roi_align_wmma<<<M, 256, 0, stream>>>(f2, f3, f4, f5, boxes, bidx, (float*)d_out);
}